// AncProbsLayer_33234456936523
// MI455X (gfx1250) — compile-verified
//
#include <hip/hip_runtime.h>
#include <hip/hip_bf16.h>

#define BB 128
#define LL 2048
#define HH 8
#define DD 20
#define NTOT 160   // H*D

typedef __attribute__((ext_vector_type(2))) float v2f;
typedef __attribute__((ext_vector_type(8))) float v8f;

__device__ __forceinline__ float softplus_f(float x) {
    // stable logaddexp(x, 0)
    return fmaxf(x, 0.0f) + log1pf(expf(-fabsf(x)));
}

// ---------------------------------------------------------------------------
// Kernel 1: per-head GTR precompute + Jacobi eigendecomposition (tiny).
// One thread per head h (8 heads). Serial 20x20 symmetric Jacobi.
// ---------------------------------------------------------------------------
__global__ void k_gtr_precompute(const float* __restrict__ exch,   // (H,1,D,D)
                                 const float* __restrict__ eq,     // (H,1,D)
                                 float* __restrict__ eigvals,      // (H,D)
                                 float* __restrict__ eigvecs,      // (H,D,D) V[d][k]
                                 float* __restrict__ sqrtp,        // (H,D)
                                 float* __restrict__ invsqrtp)     // (H,D)
{
    int h = threadIdx.x;
    if (h >= HH) return;

    const float* K = exch + h * DD * DD;
    const float* E = eq + h * DD;

    float S[DD * DD];
    float V[DD * DD];
    float p[DD];
    float diag[DD];

    // R = 0.5*(softplus(K) + softplus(K^T))  (store into S temporarily)
    for (int d = 0; d < DD; ++d)
        for (int e = 0; e < DD; ++e)
            S[d * DD + e] = 0.5f * (softplus_f(K[d * DD + e]) + softplus_f(K[e * DD + d]));

    // p = softmax(E)
    float mx = -1e30f;
    for (int d = 0; d < DD; ++d) mx = fmaxf(mx, E[d]);
    float sum = 0.0f;
    for (int d = 0; d < DD; ++d) { p[d] = expf(E[d] - mx); sum += p[d]; }
    for (int d = 0; d < DD; ++d) p[d] /= sum;

    // Q = R * p_j ; diag_d = rowsum ; Q_dd -= diag_d ; Q /= max(mue, eps)
    for (int d = 0; d < DD; ++d) {
        float rs = 0.0f;
        for (int e = 0; e < DD; ++e) {
            float q = S[d * DD + e] * p[e];
            S[d * DD + e] = q;
            rs += q;
        }
        diag[d] = rs;
    }
    float mue = 0.0f;
    for (int d = 0; d < DD; ++d) mue += p[d] * diag[d];
    float inv_mue = 1.0f / fmaxf(mue, 1e-16f);
    for (int d = 0; d < DD; ++d) S[d * DD + d] -= diag[d];
    for (int i = 0; i < DD * DD; ++i) S[i] *= inv_mue;

    // Symmetrize with sqrt(pi): Ssym = diag(sp) Q diag(1/sp), then 0.5*(S+S^T)
    float sp[DD], isp[DD];
    for (int d = 0; d < DD; ++d) { sp[d] = sqrtf(p[d]); isp[d] = 1.0f / sp[d]; }
    for (int d = 0; d < DD; ++d)
        for (int e = 0; e < DD; ++e)
            S[d * DD + e] = sp[d] * S[d * DD + e] * isp[e];
    for (int d = 0; d < DD; ++d)
        for (int e = d + 1; e < DD; ++e) {
            float v = 0.5f * (S[d * DD + e] + S[e * DD + d]);
            S[d * DD + e] = v;
            S[e * DD + d] = v;
        }

    // Jacobi: V = I
    for (int i = 0; i < DD * DD; ++i) V[i] = 0.0f;
    for (int d = 0; d < DD; ++d) V[d * DD + d] = 1.0f;

    for (int sweep = 0; sweep < 30; ++sweep) {
        float off = 0.0f;
        for (int a = 0; a < DD; ++a)
            for (int b2 = a + 1; b2 < DD; ++b2)
                off += fabsf(S[a * DD + b2]);
        if (off < 1e-12f) break;
        for (int a = 0; a < DD; ++a) {
            for (int b2 = a + 1; b2 < DD; ++b2) {
                float apq = S[a * DD + b2];
                if (fabsf(apq) < 1e-14f) continue;
                float app = S[a * DD + a];
                float aqq = S[b2 * DD + b2];
                float th = (aqq - app) / (2.0f * apq);
                float t = (th >= 0.0f ? 1.0f : -1.0f) /
                          (fabsf(th) + sqrtf(1.0f + th * th));
                float c = 1.0f / sqrtf(1.0f + t * t);
                float s = t * c;
                // columns a, b2:  A <- A J
                for (int i = 0; i < DD; ++i) {
                    float x = S[i * DD + a], y = S[i * DD + b2];
                    S[i * DD + a]  = c * x - s * y;
                    S[i * DD + b2] = s * x + c * y;
                }
                // rows a, b2:  A <- J^T A
                for (int i = 0; i < DD; ++i) {
                    float x = S[a * DD + i], y = S[b2 * DD + i];
                    S[a * DD + i]  = c * x - s * y;
                    S[b2 * DD + i] = s * x + c * y;
                }
                // accumulate eigenvectors: V <- V J (columns are eigenvectors)
                for (int i = 0; i < DD; ++i) {
                    float x = V[i * DD + a], y = V[i * DD + b2];
                    V[i * DD + a]  = c * x - s * y;
                    V[i * DD + b2] = s * x + c * y;
                }
            }
        }
    }

    for (int d = 0; d < DD; ++d) {
        eigvals[h * DD + d] = S[d * DD + d];
        sqrtp[h * DD + d] = sp[d];
        invsqrtp[h * DD + d] = isp[d];
    }
    for (int i = 0; i < DD * DD; ++i) eigvecs[h * DD * DD + i] = V[i];
}

// ---------------------------------------------------------------------------
// Kernel 2: per (b,h) build P(tau) and scatter into Pbig[b][n=h*20+z][d]
// (k-innermost layout so kernel 3 gets contiguous 8B WMMA B-fragments).
// ---------------------------------------------------------------------------
__global__ void __launch_bounds__(256)
k_build_P(const int* __restrict__ rate_idx,      // (B,H)
          const float* __restrict__ tau_kernel,  // (RATES,H,1)
          const float* __restrict__ eigvals,
          const float* __restrict__ eigvecs,
          const float* __restrict__ sqrtp,
          const float* __restrict__ invsqrtp,
          float* __restrict__ Pbig)              // (B, 160, 20)
{
    int bh = blockIdx.x;
    int b = bh / HH, h = bh % HH;
    int tid = threadIdx.x;

    __shared__ float Vs[DD * DD];
    __shared__ float es[DD];
    __shared__ float sps[DD];
    __shared__ float isps[DD];

    // tau (recomputed by every thread; trivially cheap, avoids extra sync)
    int ri = rate_idx[b * HH + h];
    float tk = tau_kernel[(size_t)ri * HH + h];
    tk = fminf(fmaxf(tk, -80.0f), 80.0f);
    float tau = softplus_f(tk);

    for (int i = tid; i < DD * DD; i += blockDim.x)
        Vs[i] = eigvecs[h * DD * DD + i];
    for (int i = tid; i < DD; i += blockDim.x) {
        es[i] = expf(eigvals[h * DD + i] * tau);
        sps[i] = sqrtp[h * DD + i];
        isps[i] = invsqrtp[h * DD + i];
    }
    __syncthreads();

    for (int idx = tid; idx < DD * DD; idx += blockDim.x) {
        int d = idx / DD, z = idx % DD;
        float acc = 0.0f;
#pragma unroll
        for (int k = 0; k < DD; ++k)
            acc += Vs[d * DD + k] * es[k] * Vs[z * DD + k];
        float Pv = isps[d] * acc * sps[z];
        Pbig[((size_t)b * NTOT + h * DD + z) * DD + d] = Pv;
    }
}

// ---------------------------------------------------------------------------
// Kernel 3: the heavy GEMM. Per b: out[2048,160] = seq[2048,20] x Pbig[20,160]
// 8 waves per WG, each wave owns a 16-row M-tile x full N=160.
// WMMA f32 16x16x4: K = 20 = 5 steps, N = 160 = 10 tiles.
// ---------------------------------------------------------------------------
__global__ void __launch_bounds__(256)
k_ancprobs_gemm(const float* __restrict__ seq,   // (B, L, 20)
                const float* __restrict__ Pbig,  // (B, 160, 20)  [n][k]
                float* __restrict__ out)         // (B, L, 160)
{
    __shared__ float Bs[NTOT * DD];  // 12.8 KB, [n][k]

    int b = blockIdx.y;
    int tid = threadIdx.x;

    const float* src = Pbig + (size_t)b * NTOT * DD;
    for (int i = tid; i < NTOT * DD; i += 256) Bs[i] = src[i];
    __syncthreads();

    int wave = tid >> 5;
    int lane = tid & 31;
    int lrow = lane & 15;    // M (A) / N (B,C) index within tile
    int khalf = lane >> 4;   // 0: K pair {0,1}, 1: K pair {2,3}

    int l0 = blockIdx.x * 128 + wave * 16;

    // A fragments: lane holds seq[b, l0+lrow, ks*4 + khalf*2 + {0,1}]
    const float* arow = seq + ((size_t)b * LL + (l0 + lrow)) * DD;
    v2f afrag[5];
#pragma unroll
    for (int ks = 0; ks < 5; ++ks)
        afrag[ks] = *(const v2f*)(arow + ks * 4 + khalf * 2);

    v8f acc[10];
#pragma unroll
    for (int nt = 0; nt < 10; ++nt) acc[nt] = (v8f){};

#pragma unroll
    for (int nt = 0; nt < 10; ++nt) {
        int n = nt * 16 + lrow;
        const float* bcol = &Bs[n * DD];
#pragma unroll
        for (int ks = 0; ks < 5; ++ks) {
            v2f bfrag = *(const v2f*)(bcol + ks * 4 + khalf * 2);
            acc[nt] = __builtin_amdgcn_wmma_f32_16x16x4_f32(
                false, afrag[ks], false, bfrag, (short)0, acc[nt],
                false, false);
        }
    }

    // C layout: VGPR r holds (M=r, lanes 0-15) and (M=r+8, lanes 16-31), N=lrow
    float* orow = out + ((size_t)b * LL + l0 + khalf * 8) * NTOT + lrow;
#pragma unroll
    for (int r = 0; r < 8; ++r) {
#pragma unroll
        for (int nt = 0; nt < 10; ++nt)
            orow[(size_t)r * NTOT + nt * 16] = acc[nt][r];
    }
}

// ---------------------------------------------------------------------------
extern "C" void kernel_launch(void* const* d_in, const int* in_sizes, int n_in,
                              void* d_out, int out_size, void* d_ws, size_t ws_size,
                              hipStream_t stream) {
    const float* seq        = (const float*)d_in[0];  // (B,L,1,1,D)
    const int*   rate_idx   = (const int*)d_in[1];    // (B,H)
    const float* tau_kernel = (const float*)d_in[2];  // (RATES,H,1)
    const float* exch       = (const float*)d_in[3];  // (H,1,D,D)
    const float* eq         = (const float*)d_in[4];  // (H,1,D)
    float* out = (float*)d_out;

    float* ws = (float*)d_ws;
    float* eigvals  = ws;                 // 160
    float* sqrtp    = ws + 160;           // 160
    float* invsqrtp = ws + 320;           // 160
    float* eigvecs  = ws + 480;           // 3200
    float* Pbig     = ws + 3680;          // B*160*20 = 409600  (~1.65 MB total)

    k_gtr_precompute<<<1, 32, 0, stream>>>(exch, eq, eigvals, eigvecs,
                                           sqrtp, invsqrtp);

    k_build_P<<<BB * HH, 256, 0, stream>>>(rate_idx, tau_kernel, eigvals,
                                           eigvecs, sqrtp, invsqrtp, Pbig);

    dim3 grid(LL / 128, BB);  // (16, 128)
    k_ancprobs_gemm<<<grid, 256, 0, stream>>>(seq, Pbig, out);
}